// GINConv_7438883357204
// MI455X (gfx1250) — compile-verified
//
#include <hip/hip_runtime.h>

typedef __attribute__((ext_vector_type(2))) float v2f;
typedef __attribute__((ext_vector_type(8))) float v8f;

#define NN    24000   // graph nodes
#define DF    32      // feature dim (d_in == d_out)
#define WAVES 4       // waves per block, K split 4 ways
#define KW    (NN / WAVES)   // 6000 K-columns per wave

// GINConv: out = relu( (X + A@X) @ W^T + b )
// One block = one 16-row tile of the output. 4 waves split the K reduction
// of A@X; partials are combined through LDS; wave 0 runs the 32x32 linear
// epilogue, also on the f32 WMMA path.
__global__ __launch_bounds__(128) void gin_wmma_f32_kernel(
    const float* __restrict__ A, const float* __restrict__ X,
    const float* __restrict__ W, const float* __restrict__ bias,
    float* __restrict__ out)
{
    // per-wave partial S tiles, padded row stride 34 (keeps 8B alignment,
    // avoids bank conflicts)
    __shared__ float S[WAVES][16][34];

    const int tid  = threadIdx.x;
    const int lane = tid & 31;
    const int wv   = tid >> 5;          // wave id 0..3
    const int row  = lane & 15;         // M (or N) index within tile
    const int kh   = lane >> 4;         // K-half selector (0/1) per WMMA layout
    const int r0   = blockIdx.x * 16;   // first row of this tile

    // ---------------- main reduction: S_tile = A[r0:r0+16, :] @ X --------
    const int wk0 = wv * KW;
    // A operand: lane 'row' streams row (r0+row); K pair (2*kh, 2*kh+1)
    const float* Ap = A + (size_t)(r0 + row) * NN + wk0 + 2 * kh;
    // B operand: lane 'row' holds column 'row' of X rows (k + 2*kh + {0,1})
    const float* Xp = X + (size_t)(wk0 + 2 * kh) * DF + row;

    v8f c0 = {};   // output columns 0..15
    v8f c1 = {};   // output columns 16..31

    for (int k = 0; k < KW; k += 16) {
        __builtin_prefetch(Ap + k + 1024, 0, 0);   // global_prefetch_b8, ~4KB ahead
#pragma unroll
        for (int u = 0; u < 16; u += 4) {
            v2f a = *(const v2f*)(Ap + k + u);                 // A[r0+row][k'+2kh .. +1]
            const float* xb = Xp + (size_t)(k + u) * DF;
            v2f bLo = { xb[0],  xb[DF] };                      // X[k'+2kh..][row]
            v2f bHi = { xb[16], xb[DF + 16] };                 // X[k'+2kh..][row+16]
            c0 = __builtin_amdgcn_wmma_f32_16x16x4_f32(
                     false, a, false, bLo, (short)0, c0, false, false);
            c1 = __builtin_amdgcn_wmma_f32_16x16x4_f32(
                     false, a, false, bHi, (short)0, c1, false, false);
        }
    }

    // stash partials in LDS (C layout: VGPR v -> M = v + 8*kh, N = row)
#pragma unroll
    for (int v = 0; v < 8; ++v) {
        S[wv][v + 8 * kh][row]      = c0[v];
        S[wv][v + 8 * kh][row + 16] = c1[v];
    }
    __syncthreads();

    // ---------------- epilogue: relu( (X_tile + sum_w S_w) @ W^T + b ) ---
    // scalar (SGPR) branch so wave 0 enters with EXEC all-1s (WMMA req.)
    if (__builtin_amdgcn_readfirstlane(wv) == 0) {
        v8f d0 = {};
        v8f d1 = {};
#pragma unroll
        for (int kc = 0; kc < 8; ++kc) {
            const int kk = kc * 4 + 2 * kh;        // K pair base (feature dim)
            // A operand: H[row][kk], H[row][kk+1] = X + sum of wave partials
            v2f h = *(const v2f*)(X + (size_t)(r0 + row) * DF + kk);
#pragma unroll
            for (int w = 0; w < WAVES; ++w) {
                h += *(const v2f*)&S[w][row][kk];
            }
            // B operand: B[k][n] = W[n][k]  (out = H @ W^T)
            v2f wLo = *(const v2f*)(W + (size_t)row * DF + kk);
            v2f wHi = *(const v2f*)(W + (size_t)(row + 16) * DF + kk);
            d0 = __builtin_amdgcn_wmma_f32_16x16x4_f32(
                     false, h, false, wLo, (short)0, d0, false, false);
            d1 = __builtin_amdgcn_wmma_f32_16x16x4_f32(
                     false, h, false, wHi, (short)0, d1, false, false);
        }

        const float b0 = bias[row];
        const float b1 = bias[row + 16];
#pragma unroll
        for (int v = 0; v < 8; ++v) {
            const int m = v + 8 * kh;              // C/D layout: M = v + 8*kh
            float* o = out + (size_t)(r0 + m) * DF;
            o[row]      = fmaxf(d0[v] + b0, 0.0f);
            o[row + 16] = fmaxf(d1[v] + b1, 0.0f);
        }
    }
}

extern "C" void kernel_launch(void* const* d_in, const int* in_sizes, int n_in,
                              void* d_out, int out_size, void* d_ws, size_t ws_size,
                              hipStream_t stream) {
    const float* A    = (const float*)d_in[0];  // [N, N]
    const float* X    = (const float*)d_in[1];  // [N, 32]
    const float* W    = (const float*)d_in[2];  // [32, 32] (row = out feature)
    const float* bias = (const float*)d_in[3];  // [32]
    float* out = (float*)d_out;                 // [N, 32]

    dim3 grid(NN / 16);   // 1500 row tiles
    dim3 block(128);      // 4 waves (wave32)
    gin_wmma_f32_kernel<<<grid, block, 0, stream>>>(A, X, W, bias, out);
}